// NeRFModulateT_28200755266117
// MI455X (gfx1250) — compile-verified
//
#include <hip/hip_runtime.h>
#include <hip/hip_bf16.h>

typedef float v2f __attribute__((ext_vector_type(2)));
typedef float v8f __attribute__((ext_vector_type(8)));

#define NRAYS    65536
#define NSAMP    192
#define NEPS     1e-10f
#define BLOCK    256

// Row-sum of 4 per-lane partials across a wave's 32 "rows" (one row per lane)
// using V_WMMA_F32_16X16X4_F32 with an all-ones B matrix:
//   D[m,n] = sum_k A[m,k]  (all 16 columns identical).
// A-matrix layout (32-bit A 16x4): lanes 0-15 hold {K=0,K=1} in the two VGPRs,
// lanes 16-31 hold {K=2,K=3}.  Two WMMAs cover rows 0-15 and 16-31.
// Results land in wb[0..31] (row r of this wave -> wb[r]).
__device__ __forceinline__ void wave_rowsum32_wmma(float p0, float p1, float p2, float p3,
                                                   int lane, float* wb) {
  // A1: rows = lanes 0..15
  float s2 = __shfl(p2, lane & 15, 32);          // p2 of row (lane&15)
  float s3 = __shfl(p3, lane & 15, 32);
  // A2: rows = lanes 16..31
  float t0 = __shfl(p0, (lane & 15) + 16, 32);   // p0 of row (lane&15)+16
  float t1 = __shfl(p1, (lane & 15) + 16, 32);
  const bool lo = lane < 16;
  v2f a1, a2, bones;
  a1.x = lo ? p0 : s2;  a1.y = lo ? p1 : s3;
  a2.x = lo ? t0 : p2;  a2.y = lo ? t1 : p3;
  bones.x = 1.0f; bones.y = 1.0f;
  v8f c0 = {};
  v8f d1 = __builtin_amdgcn_wmma_f32_16x16x4_f32(false, a1, false, bones, (short)0, c0, false, false);
  v8f d2 = __builtin_amdgcn_wmma_f32_16x16x4_f32(false, a2, false, bones, (short)0, c0, false, false);
  // C/D 16x16 layout: lanes 0-15 -> N=lane, M=vgpr; lanes 16-31 -> N=lane-16, M=vgpr+8.
  // Column N=0 lives in lane 0 (M=0..7) and lane 16 (M=8..15).
  if (lane == 0) {
#pragma unroll
    for (int v = 0; v < 8; ++v) { wb[v]      = d1[v]; wb[16 + v] = d2[v]; }
  } else if (lane == 16) {
#pragma unroll
    for (int v = 0; v < 8; ++v) { wb[8 + v]  = d1[v]; wb[24 + v] = d2[v]; }
  }
}

__global__ __launch_bounds__(BLOCK)
void nerf_render_kernel(const float* __restrict__ raw,
                        const float* __restrict__ z_vals,
                        const float* __restrict__ rays_d,
                        float* __restrict__ out) {
  const int ray  = blockIdx.x * BLOCK + (int)threadIdx.x;
  const int lane = threadIdx.x & 31;
  const int wid  = threadIdx.x >> 5;

  // Output layout: rgb_map[N,3] | density[N,191] | acc[N] | weights[N,192] | depth[N]
  float* out_rgb     = out;
  float* out_density = out + (size_t)NRAYS * 3;
  float* out_acc     = out_density + (size_t)NRAYS * (NSAMP - 1);
  float* out_w       = out_acc + (size_t)NRAYS;
  float* out_depth   = out_w + (size_t)NRAYS * NSAMP;

  const float4* rawp = (const float4*)raw + (size_t)ray * NSAMP;  // [s].xyzw = raw[ray,s,0..3]
  const float*  zp   = z_vals + (size_t)ray * NSAMP;
  const float*  rd   = rays_d + (size_t)ray * 3;

  const float dx = rd[0], dy = rd[1], dz = rd[2];
  const float dnorm = sqrtf(dx * dx + dy * dy + dz * dz);

  float trans = 1.0f;
  float part[4]  = {0.0f, 0.0f, 0.0f, 0.0f};   // phase-split partial sums of w   -> acc (WMMA)
  float zpart[4] = {0.0f, 0.0f, 0.0f, 0.0f};   // phase-split partial sums of w*z -> depth (WMMA)
  float r0 = 0.0f, r1 = 0.0f, r2 = 0.0f;
  float z_cur = zp[0];

  float* wrow = out_w + (size_t)ray * NSAMP;
  float* drow = out_density + (size_t)ray * (NSAMP - 1);

  for (int c = 0; c < NSAMP / 4; ++c) {
    __builtin_prefetch((const void*)(rawp + 4 * c + 32), 0, 3);  // global_prefetch_b8, 512B ahead
    float4 w4;
#pragma unroll
    for (int k = 0; k < 4; ++k) {
      const int s = 4 * c + k;
      const float4 rv = rawp[s];
      float alpha, z_next;
      if (s != NSAMP - 1) {
        z_next = zp[s + 1];
        const float dens = fmaxf(rv.w, 0.0f);     // relu(sigma)
        drow[s] = dens;                           // density output [N, S-1]
        const float dist = fabsf(z_next - z_cur) * dnorm;
        alpha = 1.0f - __expf(-dens * dist);
      } else {                                    // last sample: alpha forced to 1
        z_next = z_cur;
        alpha = 1.0f;
      }
      const float w = alpha * trans;
      part[k]  += w;
      zpart[k] += w * z_cur;
      r0 += w * __builtin_amdgcn_rcpf(1.0f + __expf(-rv.x));   // w * sigmoid(r)
      r1 += w * __builtin_amdgcn_rcpf(1.0f + __expf(-rv.y));
      r2 += w * __builtin_amdgcn_rcpf(1.0f + __expf(-rv.z));
      trans *= (1.0f + NEPS) - alpha;             // exclusive cumprod carry
      z_cur = z_next;
      ((float*)&w4)[k] = w;
    }
    ((float4*)wrow)[c] = w4;                      // B128 store, 16B-aligned
  }

  out_rgb[(size_t)ray * 3 + 0] = r0;
  out_rgb[(size_t)ray * 3 + 1] = r1;
  out_rgb[(size_t)ray * 3 + 2] = r2;

  // Finish acc_map and depth_map with WMMA row-sums (EXEC is all-ones here).
  __shared__ float red[2 * (BLOCK / 32) * 32];
  float* accbuf = red + wid * 32;
  float* depbuf = red + (BLOCK / 32) * 32 + wid * 32;
  wave_rowsum32_wmma(part[0],  part[1],  part[2],  part[3],  lane, accbuf);
  wave_rowsum32_wmma(zpart[0], zpart[1], zpart[2], zpart[3], lane, depbuf);
  __syncthreads();

  const int ray_base = blockIdx.x * BLOCK + wid * 32;  // == ray - lane
  out_acc[ray_base + lane]   = accbuf[lane];
  out_depth[ray_base + lane] = depbuf[lane];
}

extern "C" void kernel_launch(void* const* d_in, const int* in_sizes, int n_in,
                              void* d_out, int out_size, void* d_ws, size_t ws_size,
                              hipStream_t stream) {
  (void)in_sizes; (void)n_in; (void)d_ws; (void)ws_size; (void)out_size;
  const float* raw    = (const float*)d_in[0];   // [65536, 192, 4] f32
  const float* z_vals = (const float*)d_in[1];   // [65536, 192]    f32
  const float* rays_d = (const float*)d_in[2];   // [65536, 3]      f32
  float* out = (float*)d_out;                    // 65536*388 f32, concatenated outputs

  dim3 grid(NRAYS / BLOCK);
  dim3 block(BLOCK);
  nerf_render_kernel<<<grid, block, 0, stream>>>(raw, z_vals, rays_d, out);
}